// HybridGraphClassifier_65481071404084
// MI455X (gfx1250) — compile-verified
//
#include <hip/hip_runtime.h>
#include <cstddef>

typedef __attribute__((ext_vector_type(2))) float v2f;
typedef __attribute__((ext_vector_type(8))) float v8f;

#define D_FEAT 128
#define TILE_M 16
#define LDA_PITCH 132   // dwords; bank stride 4 -> conflict-free column reads

// ---------------- zero workspace ----------------
__global__ __launch_bounds__(256) void k_zero4(float4* __restrict__ p, int n4) {
  int i = blockIdx.x * blockDim.x + threadIdx.x;
  int stride = gridDim.x * blockDim.x;
  float4 z; z.x = 0.f; z.y = 0.f; z.z = 0.f; z.w = 0.f;
  for (; i < n4; i += stride) p[i] = z;
}

// ------------- edge-parallel scatter-add (one wave32 per edge) -------------
__global__ __launch_bounds__(256) void k_scatter(const float* __restrict__ feat,
                                                 const int* __restrict__ src,
                                                 const int* __restrict__ dst,
                                                 float* __restrict__ agg,
                                                 float* __restrict__ deg,
                                                 int nEdges) {
  int gid = blockIdx.x * blockDim.x + threadIdx.x;
  int e = gid >> 5;
  if (e >= nEdges) return;
  int lane = gid & 31;
  int s = src[e];                      // same addr across wave -> broadcast
  int d = dst[e];
  const float4 v = reinterpret_cast<const float4*>(feat + (size_t)s * D_FEAT)[lane];
  float* ap = agg + (size_t)d * D_FEAT + lane * 4;
  atomicAdd(ap + 0, v.x);
  atomicAdd(ap + 1, v.y);
  atomicAdd(ap + 2, v.z);
  atomicAdd(ap + 3, v.w);
  if (lane == 0) atomicAdd(deg + d, 1.0f);
}

// ------------- mean finalize (in place), fallback = own feature -------------
__global__ __launch_bounds__(256) void k_finalize(float4* __restrict__ agg4,
                                                  const float* __restrict__ deg,
                                                  const float4* __restrict__ fb4,
                                                  int n4) {
  int i = blockIdx.x * blockDim.x + threadIdx.x;
  if (i >= n4) return;
  int node = i >> 5;                   // (i*4)/128
  float d = deg[node];
  if (d > 0.0f) {
    float r = 1.0f / d;
    float4 a = agg4[i];
    a.x *= r; a.y *= r; a.z *= r; a.w *= r;
    agg4[i] = a;
  } else {
    agg4[i] = fb4[i];
  }
}

// ------------- out = tanh(A[M,128] @ W[128,128] + b) via V_WMMA_F32_16X16X4_F32 -------------
__global__ __launch_bounds__(256) void k_gemm_tanh(const float* __restrict__ A,
                                                   const float* __restrict__ W,
                                                   const float* __restrict__ bias,
                                                   float* __restrict__ out,
                                                   int M) {
  __shared__ float lA[TILE_M * LDA_PITCH];

  const int tid = threadIdx.x;
  const int m0  = blockIdx.x * TILE_M;

  // Stage 16x128 fp32 A tile into LDS (512 float4, 2 per thread, coalesced).
  for (int i = tid; i < 512; i += 256) {
    int row = i >> 5;          // 0..15
    int c4  = i & 31;          // float4 column
    float4 v;
    if (m0 + row < M) {
      v = reinterpret_cast<const float4*>(A + (size_t)(m0 + row) * D_FEAT)[c4];
    } else {
      v.x = 0.f; v.y = 0.f; v.z = 0.f; v.w = 0.f;
    }
    reinterpret_cast<float4*>(&lA[row * LDA_PITCH + c4 * 4])[0] = v;   // 16B aligned (132%4==0)
  }
  __syncthreads();

  const int lane  = tid & 31;
  const int wv    = tid >> 5;          // N-tile 0..7 (8 waves cover N=128)
  const int mrow  = lane & 15;         // A row held by this lane
  const int khalf = lane >> 4;         // K pair select (ISA A/B fragment layout)
  const int n     = wv * TILE_M + (lane & 15);

  const float* lArow = &lA[mrow * LDA_PITCH + khalf * 2];
  const float* Wp    = W + n;          // W row-major [K=128][N=128]

  v8f acc = {};
#pragma unroll
  for (int kk = 0; kk < 32; ++kk) {
    const int kb = kk * 4 + khalf * 2;
    v2f a, b;
    a[0] = lArow[kk * 4 + 0];          // A[mrow][kb]
    a[1] = lArow[kk * 4 + 1];          // A[mrow][kb+1]
    b[0] = Wp[(size_t)kb * D_FEAT];          // B[kb][n]
    b[1] = Wp[(size_t)(kb + 1) * D_FEAT];    // B[kb+1][n]
    acc = __builtin_amdgcn_wmma_f32_16x16x4_f32(
        /*neg_a=*/false, a, /*neg_b=*/false, b,
        /*c_mod=*/(short)0, acc, /*reuse_a=*/false, /*reuse_b=*/false);
  }

  const float bn = bias[n];
#pragma unroll
  for (int i = 0; i < 8; ++i) {
    int m = m0 + i + 8 * khalf;        // C/D layout: VGPR i -> row i (+8 for upper half-wave)
    if (m < M) out[(size_t)m * D_FEAT + n] = tanhf(acc[i] + bn);
  }
}

// ------------- head: out[M,10] = h @ W3[128,10] + b3 (tiny; VALU) -------------
__global__ __launch_bounds__(256) void k_head(const float* __restrict__ h,
                                              const float* __restrict__ W3,
                                              const float* __restrict__ b3,
                                              float* __restrict__ out,
                                              int total) {
  int t = blockIdx.x * blockDim.x + threadIdx.x;
  if (t >= total) return;
  int node = t / 10;
  int c = t - node * 10;
  const float* hp = h + (size_t)node * D_FEAT;
  float s = b3[c];
#pragma unroll 8
  for (int k = 0; k < D_FEAT; ++k) s = fmaf(hp[k], W3[k * 10 + c], s);
  out[t] = s;
}

extern "C" void kernel_launch(void* const* d_in, const int* in_sizes, int n_in,
                              void* d_out, int out_size, void* d_ws, size_t ws_size,
                              hipStream_t stream) {
  const float* x    = (const float*)d_in[0];
  const int*   esrc = (const int*)d_in[1];
  const int*   edst = (const int*)d_in[2];
  const float* W1   = (const float*)d_in[3];
  const float* b1   = (const float*)d_in[4];
  const float* W2   = (const float*)d_in[5];
  const float* b2   = (const float*)d_in[6];
  const float* W3   = (const float*)d_in[7];
  const float* b3   = (const float*)d_in[8];
  float* out = (float*)d_out;

  const int M = in_sizes[0] / D_FEAT;        // 100000
  const int E = in_sizes[1];                 // 1600000
  const size_t NODE_F = (size_t)M * D_FEAT;

  // workspace layout: [deg (padded 102400 f32)][agg (M*128)][h (M*128)]
  float* deg = (float*)d_ws;
  const size_t DEG_PAD = 102400;
  float* agg = deg + DEG_PAD;
  float* h   = agg + NODE_F;

  const dim3 blk(256);
  const int zero_n4   = (int)((DEG_PAD + NODE_F) / 4);     // deg+agg are contiguous
  const int zero_grid = (zero_n4 + 255) / 256;
  const int scat_grid = (int)(((size_t)E * 32 + 255) / 256);
  const int fin_n4    = (int)(NODE_F / 4);
  const int fin_grid  = (fin_n4 + 255) / 256;
  const int gemm_grid = (M + TILE_M - 1) / TILE_M;         // 6250
  const int head_tot  = M * 10;
  const int head_grid = (head_tot + 255) / 256;

  // ---- layer 1: aggregate(x) -> tanh(.@W1 + b1) ----
  k_zero4<<<zero_grid, blk, 0, stream>>>((float4*)d_ws, zero_n4);
  k_scatter<<<scat_grid, blk, 0, stream>>>(x, esrc, edst, agg, deg, E);
  k_finalize<<<fin_grid, blk, 0, stream>>>((float4*)agg, deg, (const float4*)x, fin_n4);
  k_gemm_tanh<<<gemm_grid, blk, 0, stream>>>(agg, W1, b1, h, M);

  // ---- layer 2: aggregate(h) -> tanh(.@W2 + b2) ----
  k_zero4<<<zero_grid, blk, 0, stream>>>((float4*)d_ws, zero_n4);
  k_scatter<<<scat_grid, blk, 0, stream>>>(h, esrc, edst, agg, deg, E);
  k_finalize<<<fin_grid, blk, 0, stream>>>((float4*)agg, deg, (const float4*)h, fin_n4);
  k_gemm_tanh<<<gemm_grid, blk, 0, stream>>>(agg, W2, b2, h, M);

  // ---- head ----
  k_head<<<head_grid, blk, 0, stream>>>(h, W3, b3, out, head_tot);
}